// AttentionModulator_45346264711386
// MI455X (gfx1250) — compile-verified
//
#include <hip/hip_runtime.h>
#include <cstdint>

// Problem constants (fixed by the reference).
#define B_  2
#define H_  16
#define Q_  1024
#define K_  2048
#define NIDS 64
#define ROWS_TOTAL      (B_ * H_ * Q_)   // 32768 softmax rows
#define ROWS_PER_BLOCK  16
#define THREADS         256              // 8 wave32s
#define VEC_PER_THREAD  2                // 2 x vec4 = 8 floats/thread; 256*8 = K_

// Native clang vector type: legal for __builtin_nontemporal_store and still
// lowers to b128 global/LDS accesses.
typedef float v4f __attribute__((ext_vector_type(4)));

// ---------------------------------------------------------------------------
// CDNA5 async global -> LDS DMA (ASYNCcnt-tracked, CDNA5 ISA).
// vdst VGPR = LDS byte offset (workgroup-relative), vaddr = 64-bit global addr.
// TH_LOAD_NT: this data is consumed exactly once from LDS — keep it out of
// the 192MB L2's working set so `scale` (reused by all blocks) stays resident.
// ---------------------------------------------------------------------------
__device__ __forceinline__ void async_ld_b128_nt(unsigned lds_off,
                                                 unsigned long long gaddr) {
  asm volatile("global_load_async_to_lds_b128 %0, %1, off th:TH_LOAD_NT"
               :
               : "v"(lds_off), "v"(gaddr)
               : "memory");
}
__device__ __forceinline__ void wait_async_le4() {
  asm volatile("s_wait_asynccnt 0x4" ::: "memory");
}
__device__ __forceinline__ void wait_async_0() {
  asm volatile("s_wait_asynccnt 0x0" ::: "memory");
}

// Low 32 bits of a generic pointer into __shared__ == LDS byte offset.
__device__ __forceinline__ unsigned lds_off_of(const void* p) {
  return (unsigned)(uintptr_t)p;
}

// Wave32 reductions.
__device__ __forceinline__ float wave_max(float v) {
#pragma unroll
  for (int o = 16; o > 0; o >>= 1) v = fmaxf(v, __shfl_xor(v, o, 32));
  return v;
}
__device__ __forceinline__ float wave_sum(float v) {
#pragma unroll
  for (int o = 16; o > 0; o >>= 1) v += __shfl_xor(v, o, 32);
  return v;
}

// ---------------------------------------------------------------------------
// Pass 1: scale[b*K + k] = (isin(id, hyper)?1.18:1) * (isin(id, avoid)?0.999:1)
// ---------------------------------------------------------------------------
__global__ __launch_bounds__(THREADS)
void build_scale_kernel(const int* __restrict__ input_ids,
                        const int* __restrict__ hyper_ids,
                        const int* __restrict__ avoid_ids,
                        float* __restrict__ scale, int n) {
  __shared__ int shH[NIDS];
  __shared__ int shA[NIDS];
  const int t = threadIdx.x;
  if (t < NIDS)            shH[t]        = hyper_ids[t];
  else if (t < 2 * NIDS)   shA[t - NIDS] = avoid_ids[t - NIDS];
  __syncthreads();

  for (int i = blockIdx.x * blockDim.x + t; i < n; i += gridDim.x * blockDim.x) {
    const int id = input_ids[i];
    bool h = false, a = false;
#pragma unroll 8
    for (int j = 0; j < NIDS; ++j) {
      h = h || (id == shH[j]);
      a = a || (id == shA[j]);
    }
    const float s = (h ? (1.0f + 1.8f * 0.1f) : 1.0f) *
                    (a ? (1.0f - 0.01f * 0.1f) : 1.0f);
    scale[i] = s;
  }
}

// ---------------------------------------------------------------------------
// Pass 2: streaming modulated softmax. One block = 16 rows (same b), double-
// buffered async DMA of attn/noise rows into LDS, wave32 tree reductions,
// non-temporal output stores.
// ---------------------------------------------------------------------------
__global__ __launch_bounds__(THREADS)
void softmax_mod_kernel(const float* __restrict__ attn,
                        const float* __restrict__ noise,
                        const float* __restrict__ scale,
                        float* __restrict__ out) {
  __shared__ float sA[2][K_];          // 16 KB: attn row, double buffered
  __shared__ float sN[2][K_];          // 16 KB: noise row, double buffered
  __shared__ float sScale[K_];         //  8 KB: per-(b,k) scale row
  __shared__ float sRed[THREADS / 32]; // cross-wave reduction slots

  const int tid  = threadIdx.x;
  const int lane = tid & 31;
  const int wid  = tid >> 5;
  const int row0 = blockIdx.x * ROWS_PER_BLOCK;
  const int b    = row0 / (H_ * Q_);   // constant within the block

  // Stage the scale row for this batch once per block (RT policy: this 16KB
  // array is re-read by every block and should live in L2).
#pragma unroll
  for (int i = 0; i < VEC_PER_THREAD; ++i) {
    const int v = tid + i * THREADS;   // vec4 index
    *reinterpret_cast<v4f*>(&sScale[v * 4]) =
        *reinterpret_cast<const v4f*>(&scale[(size_t)b * K_ + (size_t)v * 4]);
  }
  __syncthreads();

  // Issue the 4 async b128 DMAs (per wave) that fill buffer s with row `row`.
  auto issue_row = [&](int s, int row) {
    const float* ag = attn  + (size_t)row * K_;
    const float* ng = noise + (size_t)row * K_;
#pragma unroll
    for (int i = 0; i < VEC_PER_THREAD; ++i) {
      const int v = tid + i * THREADS;
      async_ld_b128_nt(lds_off_of(&sA[s][v * 4]),
                       (unsigned long long)(uintptr_t)(ag + (size_t)v * 4));
      async_ld_b128_nt(lds_off_of(&sN[s][v * 4]),
                       (unsigned long long)(uintptr_t)(ng + (size_t)v * 4));
    }
  };

  issue_row(0, row0);  // prime the pipeline

  for (int j = 0; j < ROWS_PER_BLOCK; ++j) {
    const int  s    = j & 1;
    const bool more = (j + 1) < ROWS_PER_BLOCK;
    if (more) issue_row(s ^ 1, row0 + j + 1);  // prefetch next row
    // Async loads complete in order: waiting to <=4 releases exactly the
    // 4 older (current-buffer) DMAs while the prefetch stays in flight.
    if (more) wait_async_le4();
    else      wait_async_0();

    // x = (attn + 0.1*noise) * scale ; local max
    float x[VEC_PER_THREAD * 4];
    float m = -3.402823466e+38f;
#pragma unroll
    for (int i = 0; i < VEC_PER_THREAD; ++i) {
      const int v = tid + i * THREADS;
      const v4f a  = *reinterpret_cast<const v4f*>(&sA[s][v * 4]);
      const v4f nz = *reinterpret_cast<const v4f*>(&sN[s][v * 4]);
      const v4f sc = *reinterpret_cast<const v4f*>(&sScale[v * 4]);
#pragma unroll
      for (int c = 0; c < 4; ++c) {
        x[i * 4 + c] = fmaf(0.1f, nz[c], a[c]) * sc[c];
        m = fmaxf(m, x[i * 4 + c]);
      }
    }

    // Row max across 256 threads.
    m = wave_max(m);
    __syncthreads();                 // protect sRed reuse across rows
    if (lane == 0) sRed[wid] = m;
    __syncthreads();
    float bm = sRed[0];
#pragma unroll
    for (int w = 1; w < THREADS / 32; ++w) bm = fmaxf(bm, sRed[w]);

    // exp + row sum.
    float sum = 0.0f;
#pragma unroll
    for (int e = 0; e < VEC_PER_THREAD * 4; ++e) {
      x[e] = __expf(x[e] - bm);      // v_exp_f32 (TRANS)
      sum += x[e];
    }
    sum = wave_sum(sum);
    __syncthreads();
    if (lane == 0) sRed[wid] = sum;
    __syncthreads();
    float bs = 0.0f;
#pragma unroll
    for (int w = 0; w < THREADS / 32; ++w) bs += sRed[w];
    const float inv = 1.0f / bs;

    // Normalize and store non-temporally (output is never re-read; don't
    // let 256MB of writes rinse through L2).
    float* orow = out + (size_t)(row0 + j) * K_;
#pragma unroll
    for (int i = 0; i < VEC_PER_THREAD; ++i) {
      const int v = tid + i * THREADS;
      v4f r;
      r[0] = x[i * 4 + 0] * inv;
      r[1] = x[i * 4 + 1] * inv;
      r[2] = x[i * 4 + 2] * inv;
      r[3] = x[i * 4 + 3] * inv;
      __builtin_nontemporal_store(r, reinterpret_cast<v4f*>(&orow[(size_t)v * 4]));
    }
  }
}

// ---------------------------------------------------------------------------
extern "C" void kernel_launch(void* const* d_in, const int* in_sizes, int n_in,
                              void* d_out, int out_size, void* d_ws, size_t ws_size,
                              hipStream_t stream) {
  (void)in_sizes; (void)n_in; (void)out_size; (void)ws_size;

  const float* attn      = (const float*)d_in[0];
  const float* noise     = (const float*)d_in[1];
  const int*   input_ids = (const int*)d_in[2];   // [B,K]
  const int*   hyper_ids = (const int*)d_in[3];   // [64]
  const int*   avoid_ids = (const int*)d_in[4];   // [64]
  float*       out       = (float*)d_out;
  float*       scale     = (float*)d_ws;          // B*K floats = 16 KB scratch

  build_scale_kernel<<<16, THREADS, 0, stream>>>(input_ids, hyper_ids, avoid_ids,
                                                 scale, B_ * K_);
  softmax_mod_kernel<<<ROWS_TOTAL / ROWS_PER_BLOCK, THREADS, 0, stream>>>(
      attn, noise, scale, out);
}